// RecurrentGCN_74655121539736
// MI455X (gfx1250) — compile-verified
//
#include <hip/hip_runtime.h>
#include <hip/hip_bf16.h>

typedef __attribute__((ext_vector_type(2))) float v2f;
typedef __attribute__((ext_vector_type(8))) float v8f;

// ---------------------------------------------------------------------------
// Degree accumulation: deg_out[src] += w ; deg_in[dst] += w
// ---------------------------------------------------------------------------
__global__ void deg_kernel(const int* __restrict__ src, const int* __restrict__ dst,
                           const float* __restrict__ ew,
                           float* __restrict__ dego, float* __restrict__ degi, int E) {
    int e = blockIdx.x * blockDim.x + threadIdx.x;
    if (e >= E) return;
    float w = ew[e];
    atomicAdd(&dego[src[e]], w);
    atomicAdd(&degi[dst[e]], w);
}

// ---------------------------------------------------------------------------
// Edge normalization: nf = w/deg_out[src], nb = w/deg_in[dst]; also writes A out
// ---------------------------------------------------------------------------
__global__ void norm_kernel(const int* __restrict__ src, const int* __restrict__ dst,
                            const float* __restrict__ ew,
                            const float* __restrict__ dego, const float* __restrict__ degi,
                            float* __restrict__ nf, float* __restrict__ nb,
                            float* __restrict__ outA_f, float* __restrict__ outA_b, int E) {
    int e = blockIdx.x * blockDim.x + threadIdx.x;
    if (e >= E) return;
    float w = ew[e];
    float d0 = dego[src[e]]; d0 = (d0 == 0.0f) ? 1.0f : d0;
    float d1 = degi[dst[e]]; d1 = (d1 == 0.0f) ? 1.0f : d1;
    float f = w / d0, b = w / d1;
    nf[e] = f; nb[e] = b;
    outA_f[e] = f; outA_b[e] = b;
}

// ---------------------------------------------------------------------------
// MLP preprocess: h = prelu(prelu(x@w1+b1,a1)@w2+b2,a2). Weights staged in LDS.
// ---------------------------------------------------------------------------
__global__ __launch_bounds__(256) void mlp_kernel(
    const float* __restrict__ x, const float* __restrict__ w1, const float* __restrict__ b1,
    const float* __restrict__ a1, const float* __restrict__ w2, const float* __restrict__ b2,
    const float* __restrict__ a2, float* __restrict__ hout, int n) {
    __shared__ float sw1[64 * 100];
    __shared__ float sw2[100 * 64];
    __shared__ float sb1[100];
    __shared__ float sb2[64];
    for (int i = threadIdx.x; i < 6400; i += 256) { sw1[i] = w1[i]; sw2[i] = w2[i]; }
    for (int i = threadIdx.x; i < 100; i += 256) sb1[i] = b1[i];
    for (int i = threadIdx.x; i < 64; i += 256) sb2[i] = b2[i];
    __syncthreads();
    int node = blockIdx.x * 256 + threadIdx.x;
    if (node >= n) return;
    const float alpha1 = a1[0], alpha2 = a2[0];
    float xr[64];
    const float* xrow = x + (size_t)node * 64;
    #pragma unroll
    for (int i = 0; i < 64; ++i) xr[i] = xrow[i];
    float acc[64];
    #pragma unroll
    for (int c = 0; c < 64; ++c) acc[c] = sb2[c];
    for (int j = 0; j < 100; ++j) {
        float h = sb1[j];
        #pragma unroll
        for (int i = 0; i < 64; ++i) h += xr[i] * sw1[i * 100 + j];
        h = (h >= 0.0f) ? h : alpha1 * h;   // LDS broadcast reads: conflict-free
        #pragma unroll
        for (int c = 0; c < 64; ++c) acc[c] += h * sw2[j * 64 + c];
    }
    float* hrow = hout + (size_t)node * 64;
    #pragma unroll
    for (int c = 0; c < 64; ++c) {
        float v = acc[c];
        hrow[c] = (v >= 0.0f) ? v : alpha2 * v;
    }
}

// ---------------------------------------------------------------------------
// XH = concat(X, H) (H==0 when h_zero)
// ---------------------------------------------------------------------------
__global__ void concat_kernel(const float* __restrict__ X, const float* __restrict__ H,
                              int h_zero, float* __restrict__ XH, int n) {
    size_t i = (size_t)blockIdx.x * blockDim.x + threadIdx.x;
    if (i >= (size_t)n * 128) return;
    size_t row = i >> 7;
    int col = (int)(i & 127);
    float v;
    if (col < 64) v = X[row * 64 + col];
    else          v = h_zero ? 0.0f : H[row * 64 + (col - 64)];
    XH[i] = v;
}

// ---------------------------------------------------------------------------
// Chebyshev init: P2F = P2B = -M   (so x2 = 2*P(x1) - x0 via scaled scatter)
// ---------------------------------------------------------------------------
__global__ void neginit_kernel(const float* __restrict__ M,
                               float* __restrict__ P2F, float* __restrict__ P2B, int n) {
    size_t i = (size_t)blockIdx.x * blockDim.x + threadIdx.x;
    if (i >= (size_t)n * 128) return;
    float v = -M[i];
    P2F[i] = v;
    P2B[i] = v;
}

// ---------------------------------------------------------------------------
// Diffusion hop (SpMM scatter): out[sidx[e]] += scale*nrm[e]*in[gidx[e]]
// One wave per edge: 32 lanes x float4 = 128 features, coalesced gather.
// ---------------------------------------------------------------------------
__global__ void prop_kernel(const float* __restrict__ in, float* __restrict__ out,
                            const int* __restrict__ gidx, const int* __restrict__ sidx,
                            const float* __restrict__ nrm, float scale, int E) {
    long t = (long)blockIdx.x * blockDim.x + threadIdx.x;
    long e = t >> 5;
    if (e >= E) return;
    int lane = (int)(t & 31);
    float w = scale * nrm[e];
    const float4* srow = (const float4*)(in + (size_t)gidx[e] * 128);
    float* drow = out + (size_t)sidx[e] * 128;
    float4 v = srow[lane];
    atomicAdd(drow + lane * 4 + 0, w * v.x);
    atomicAdd(drow + lane * 4 + 1, w * v.y);
    atomicAdd(drow + lane * 4 + 2, w * v.z);
    atomicAdd(drow + lane * 4 + 3, w * v.w);
}

// ---------------------------------------------------------------------------
// Fused Z/R gate GEMM via V_WMMA_F32_16X16X4_F32.
// Block = 128 thr (4 waves); block -> 16 rows; wave w -> cols [16w,16w+16).
// H = sum_t mats[t] @ W[d_t,k_t]; Z=sigmoid(H_z+bz), R=sigmoid(H_r+br).
// Epilogue writes Z and XRH = [Xin | R*H].
// ---------------------------------------------------------------------------
__global__ __launch_bounds__(128) void zr_gemm_kernel(
    const float* __restrict__ XH, const float* __restrict__ P1F, const float* __restrict__ P1B,
    const float* __restrict__ P2F, const float* __restrict__ P2B,
    const float* __restrict__ WZ, const float* __restrict__ WR,
    const float* __restrict__ BZ, const float* __restrict__ BR,
    const float* __restrict__ Xin, const float* __restrict__ H, int h_zero,
    float* __restrict__ Zbuf, float* __restrict__ XRH, int n) {
    const int r0 = blockIdx.x * 16;
    const int wave = threadIdx.x >> 5;
    const int lane = threadIdx.x & 31;
    const int c0 = wave * 16;
    const int mrow = lane & 15;
    const int khalf = (lane >> 4) * 2;            // A/B fragment K phase (0 or 2)
    int arow = r0 + mrow; if (arow >= n) arow = n - 1;   // clamp, keep EXEC all-ones

    v8f accZ = {};
    v8f accR = {};
    const float* mats[6] = {XH, XH, P1F, P1B, P2F, P2B};
    const int woff[6] = {0, 3, 1, 4, 2, 5};       // (d,k) -> d*3+k for terms
    #pragma unroll
    for (int t = 0; t < 6; ++t) {
        const float* Arow = mats[t] + (size_t)arow * 128;
        const float* wz_t = WZ + (size_t)woff[t] * 128 * 64;
        const float* wr_t = WR + (size_t)woff[t] * 128 * 64;
        for (int kb = 0; kb < 128; kb += 4) {
            const int k = kb + khalf;
            v2f a;  a.x = Arow[k];                a.y = Arow[k + 1];
            v2f bz; bz.x = wz_t[k * 64 + c0 + mrow]; bz.y = wz_t[(k + 1) * 64 + c0 + mrow];
            v2f br; br.x = wr_t[k * 64 + c0 + mrow]; br.y = wr_t[(k + 1) * 64 + c0 + mrow];
            accZ = __builtin_amdgcn_wmma_f32_16x16x4_f32(false, a, false, bz,
                                                         (short)0, accZ, false, false);
            accR = __builtin_amdgcn_wmma_f32_16x16x4_f32(false, a, false, br,
                                                         (short)0, accR, false, false);
        }
    }
    const int colg = c0 + mrow;
    const float bzv = BZ[colg], brv = BR[colg];
    #pragma unroll
    for (int r = 0; r < 8; ++r) {
        int row = r0 + r + 8 * (lane >> 4);
        if (row < n) {
            float z = 1.0f / (1.0f + __expf(-(accZ[r] + bzv)));
            float rr = 1.0f / (1.0f + __expf(-(accR[r] + brv)));
            Zbuf[(size_t)row * 64 + colg] = z;
            float hv = h_zero ? 0.0f : H[(size_t)row * 64 + colg];
            XRH[(size_t)row * 128 + 64 + colg] = rr * hv;
        }
    }
    for (int i = threadIdx.x; i < 16 * 64; i += 128) {   // XRH[:, :64] = Xin
        int row = r0 + (i >> 6), col = i & 63;
        if (row < n) XRH[(size_t)row * 128 + col] = Xin[(size_t)row * 64 + col];
    }
}

// ---------------------------------------------------------------------------
// Candidate GEMM + GRU combine: Ht=tanh(dconv(XRH)+bh); Hn = Z*H + (1-Z)*Ht
// ---------------------------------------------------------------------------
__global__ __launch_bounds__(128) void cand_gemm_kernel(
    const float* __restrict__ XRH, const float* __restrict__ P1F, const float* __restrict__ P1B,
    const float* __restrict__ P2F, const float* __restrict__ P2B,
    const float* __restrict__ WH, const float* __restrict__ BH,
    const float* __restrict__ Zbuf, const float* __restrict__ H, int h_zero,
    float* __restrict__ Hout, int n) {
    const int r0 = blockIdx.x * 16;
    const int wave = threadIdx.x >> 5;
    const int lane = threadIdx.x & 31;
    const int c0 = wave * 16;
    const int mrow = lane & 15;
    const int khalf = (lane >> 4) * 2;
    int arow = r0 + mrow; if (arow >= n) arow = n - 1;

    v8f acc = {};
    const float* mats[6] = {XRH, XRH, P1F, P1B, P2F, P2B};
    const int woff[6] = {0, 3, 1, 4, 2, 5};
    #pragma unroll
    for (int t = 0; t < 6; ++t) {
        const float* Arow = mats[t] + (size_t)arow * 128;
        const float* wh_t = WH + (size_t)woff[t] * 128 * 64;
        for (int kb = 0; kb < 128; kb += 4) {
            const int k = kb + khalf;
            v2f a; a.x = Arow[k];                a.y = Arow[k + 1];
            v2f b; b.x = wh_t[k * 64 + c0 + mrow]; b.y = wh_t[(k + 1) * 64 + c0 + mrow];
            acc = __builtin_amdgcn_wmma_f32_16x16x4_f32(false, a, false, b,
                                                        (short)0, acc, false, false);
        }
    }
    const int colg = c0 + mrow;
    const float bhv = BH[colg];
    #pragma unroll
    for (int r = 0; r < 8; ++r) {
        int row = r0 + r + 8 * (lane >> 4);
        if (row < n) {
            float ht = tanhf(acc[r] + bhv);
            float z = Zbuf[(size_t)row * 64 + colg];
            float hv = h_zero ? 0.0f : H[(size_t)row * 64 + colg];
            Hout[(size_t)row * 64 + colg] = z * hv + (1.0f - z) * ht;
        }
    }
}

// ---------------------------------------------------------------------------
// Head: out = relu(H2 @ wl + bl)
// ---------------------------------------------------------------------------
__global__ void head_kernel(const float* __restrict__ H2, const float* __restrict__ wl,
                            const float* __restrict__ bl, float* __restrict__ out, int n) {
    int node = blockIdx.x * blockDim.x + threadIdx.x;
    if (node >= n) return;
    const float* hrow = H2 + (size_t)node * 64;
    float acc = bl[0];
    #pragma unroll
    for (int c = 0; c < 64; ++c) acc += hrow[c] * wl[c];
    out[node] = (acc > 0.0f) ? acc : 0.0f;
}

// ---------------------------------------------------------------------------
extern "C" void kernel_launch(void* const* d_in, const int* in_sizes, int n_in,
                              void* d_out, int out_size, void* d_ws, size_t ws_size,
                              hipStream_t stream) {
    const float* x  = (const float*)d_in[0];
    const int*   ei = (const int*)d_in[1];
    const float* ew = (const float*)d_in[2];
    const float* w1 = (const float*)d_in[3];
    const float* b1 = (const float*)d_in[4];
    const float* a1 = (const float*)d_in[5];
    const float* w2 = (const float*)d_in[6];
    const float* b2 = (const float*)d_in[7];
    const float* a2 = (const float*)d_in[8];
    const float* wz = (const float*)d_in[9];
    const float* bz = (const float*)d_in[10];
    const float* wr = (const float*)d_in[11];
    const float* br = (const float*)d_in[12];
    const float* wh = (const float*)d_in[13];
    const float* bh = (const float*)d_in[14];
    const float* wl = (const float*)d_in[15];
    const float* bl = (const float*)d_in[16];
    (void)n_in; (void)out_size; (void)ws_size;

    const int n = in_sizes[0] / 64;
    const int E = in_sizes[2];
    const int* src = ei;
    const int* dst = ei + E;

    float* out    = (float*)d_out;
    float* outA_f = out + n;        // A[0] = nf
    float* outA_b = out + n + E;    // A[1] = nb

    float* wsf = (float*)d_ws;
    size_t off = 0;
    auto alloc = [&](size_t cnt) { float* p = wsf + off; off += cnt; return p; };
    float* f_nf  = alloc(E);
    float* f_nb  = alloc(E);
    float* f_dego = alloc(n);
    float* f_degi = alloc(n);
    float* f_h   = alloc((size_t)n * 64);
    float* f_XH  = alloc((size_t)n * 128);
    float* f_XRH = alloc((size_t)n * 128);
    float* f_P1F = alloc((size_t)n * 128);   // P1F,P1B contiguous (one memset)
    float* f_P1B = alloc((size_t)n * 128);
    float* f_P2F = alloc((size_t)n * 128);
    float* f_P2B = alloc((size_t)n * 128);
    float* f_Z   = alloc((size_t)n * 64);
    float* f_H1  = alloc((size_t)n * 64);
    float* f_H2  = alloc((size_t)n * 64);

    const int eb   = (E + 255) / 256;
    const int nb_  = (n + 255) / 256;
    const int nk   = (int)(((size_t)n * 128 + 255) / 256);
    const int pb   = (int)(((long)E * 32 + 255) / 256);
    const int gemb = (n + 15) / 16;

    // ---- edge normalization + A output ----
    hipMemsetAsync(f_dego, 0, (size_t)2 * n * sizeof(float), stream);
    deg_kernel<<<eb, 256, 0, stream>>>(src, dst, ew, f_dego, f_degi, E);
    norm_kernel<<<eb, 256, 0, stream>>>(src, dst, ew, f_dego, f_degi,
                                        f_nf, f_nb, outA_f, outA_b, E);
    // ---- MLP preprocess ----
    mlp_kernel<<<nb_, 256, 0, stream>>>(x, w1, b1, a1, w2, b2, a2, f_h, n);

    // one bidirectional Chebyshev chain for matrix M -> P1F,P1B,P2F,P2B
    auto chain = [&](const float* M) {
        hipMemsetAsync(f_P1F, 0, (size_t)2 * n * 128 * sizeof(float), stream);
        prop_kernel<<<pb, 256, 0, stream>>>(M, f_P1F, src, dst, f_nf, 1.0f, E);
        prop_kernel<<<pb, 256, 0, stream>>>(M, f_P1B, dst, src, f_nb, 1.0f, E);
        neginit_kernel<<<nk, 256, 0, stream>>>(M, f_P2F, f_P2B, n);
        prop_kernel<<<pb, 256, 0, stream>>>(f_P1F, f_P2F, src, dst, f_nf, 2.0f, E);
        prop_kernel<<<pb, 256, 0, stream>>>(f_P1B, f_P2B, dst, src, f_nb, 2.0f, E);
    };

    auto cell = [&](const float* Xin, const float* Hin, int h_zero, float* Hout) {
        concat_kernel<<<nk, 256, 0, stream>>>(Xin, Hin, h_zero, f_XH, n);
        chain(f_XH);
        zr_gemm_kernel<<<gemb, 128, 0, stream>>>(f_XH, f_P1F, f_P1B, f_P2F, f_P2B,
                                                 wz, wr, bz, br, Xin, Hin, h_zero,
                                                 f_Z, f_XRH, n);
        chain(f_XRH);
        cand_gemm_kernel<<<gemb, 128, 0, stream>>>(f_XRH, f_P1F, f_P1B, f_P2F, f_P2B,
                                                   wh, bh, f_Z, Hin, h_zero, Hout, n);
    };

    cell(f_h, f_h, 1, f_H1);   // H0 == 0
    cell(x, f_H1, 0, f_H2);

    head_kernel<<<nb_, 256, 0, stream>>>(f_H2, wl, bl, out, n);
}